// Distance_11819749998715
// MI455X (gfx1250) — compile-verified
//
#include <hip/hip_runtime.h>
#include <math.h>

typedef __attribute__((ext_vector_type(2))) float v2f;
typedef __attribute__((ext_vector_type(8))) float v8f;

#define HH 320
#define WW 320
#define NB 4
#define KT 20   // H / 16 k-tiles
#define IT 20   // H / 16 i-tiles

// Bare CDNA5 v_min_num_f32 (skips HIP fminf's NaN-quieting canonicalize ops;
// inputs here are never NaN: squared distances + integer parabola terms).
__device__ __forceinline__ float vminf(float a, float b) {
    float d;
    asm("v_min_num_f32 %0, %1, %2" : "=v"(d) : "v"(a), "v"(b));
    return d;
}

// ---------------------------------------------------------------------------
// Stage 1: row-wise 1D squared distance (exact, two scans), stored TRANSPOSED:
//   gT[((mask*NB + b)*WW + w)*HH + h] = d1(b,h,w)^2   for mask in {0(==0),1(==255)}
// One thread per (mask,b,h) row. 2560 rows total.
// ---------------------------------------------------------------------------
__global__ void edt_rows(const int* __restrict__ tri, float* __restrict__ gT) {
    __shared__ float dl[32][HH];          // 40 KB LDS
    int t    = blockIdx.x * 32 + threadIdx.x;   // 0..2559
    int h    = t % HH;
    int rest = t / HH;
    int b    = rest % NB;
    int mask = rest / NB;
    int val  = mask ? 255 : 0;

    const int* row = tri + ((size_t)(b * HH + h)) * WW;
    float* dlr = dl[threadIdx.x];

    // left scan: distance to last zero-pixel (sentinel -1e6 like reference BIG)
    float last = -1.0e6f;
    for (int w = 0; w < WW; ++w) {
        if (row[w] == val) last = (float)w;
        dlr[w] = (float)w - last;
    }
    // right scan + min + square, write transposed
    float* gcol = gT + (size_t)(mask * NB + b) * WW * HH;
    float nxt = 1.0e6f;
    for (int w = WW - 1; w >= 0; --w) {
        if (row[w] == val) nxt = (float)w;
        float d = fminf(dlr[w], nxt - (float)w);
        gcol[(size_t)w * HH + h] = d * d;
    }
}

// ---------------------------------------------------------------------------
// Stage 2: per-column parabolic min via WMMA.
//   D2[i] = min_k g[k] + (i-k)^2,  (i-k)^2 rank-3 expansion:
//   A[i,:]=(1, i^2, i, 0)  B[:,k]=(g[k]+k^2, 1, -2k, 0)  => (A x B)[i,k] = cand
// One wave handles one (mask,b,w, 16-row i-tile): 20 WMMAs (fully unrolled,
// loads issued by all 32 lanes -> no exec divergence, same cacheline for both
// halves) + v_min accumulate, 16-lane cross-min, 3 Gaussian channels, uint8
// quantize. 51200 waves => grid 6400 x 256 (all waves full, EXEC all ones).
// ---------------------------------------------------------------------------
__global__ void edt_cols(const float* __restrict__ gT, float* __restrict__ out) {
    int lane = threadIdx.x & 31;
    int wv   = (blockIdx.x * blockDim.x + threadIdx.x) >> 5;

    int w    = wv % WW;  int rest = wv / WW;
    int it   = rest % IT; rest /= IT;
    int b    = rest % NB;
    int mask = rest / NB;

    int  r  = lane & 15;
    bool hi = lane >= 16;            // lanes 16-31 hold A/B rows K=2,3

    float fi = (float)(it * 16 + r); // output row handled by this lane's M slot
    v2f A;
    A.x = hi ? fi        : 1.0f;     // K2 = i        | K0 = 1
    A.y = hi ? 0.0f      : fi * fi;  // K3 = 0        | K1 = i^2
    float by = hi ? 0.0f : 1.0f;     // K3 = 0        | K1 = 1   (loop-invariant)

    const float* gcol = gT + ((size_t)(mask * NB + b) * WW + w) * HH;

    v8f acc;
#pragma unroll
    for (int j = 0; j < 8; ++j) acc[j] = 3.0e38f;

#pragma unroll
    for (int kt = 0; kt < KT; ++kt) {
        float fk   = (float)(kt * 16 + r);
        float gval = gcol[kt * 16 + r];      // ALL lanes load (halves share line)
        float blo  = __builtin_fmaf(fk, fk, gval);  // g[k] + k^2
        float bhi  = -2.0f * fk;                    // -2k
        v2f Bm;
        Bm.x = hi ? bhi : blo;               // cndmask, no branch
        Bm.y = by;
        v8f c = {};                          // candidates, C = 0
        c = __builtin_amdgcn_wmma_f32_16x16x4_f32(
                false, A, false, Bm, (short)0, c, false, false);
#pragma unroll
        for (int j = 0; j < 8; ++j) acc[j] = vminf(acc[j], c[j]);
    }

    // min across the 16 lanes of each half (n = k mod 16 candidates)
#pragma unroll
    for (int off = 1; off < 16; off <<= 1) {
#pragma unroll
        for (int j = 0; j < 8; ++j)
            acc[j] = vminf(acc[j], __shfl_xor(acc[j], off, 16));
    }

    // lanes 0-7 emit rows i0+0..7 (lower half), lanes 16-23 rows i0+8..15
    int j = lane & 7;
    float d2 = acc[0];
#pragma unroll
    for (int q = 1; q < 8; ++q) d2 = (j == q) ? acc[q] : d2;   // cndmask chain

    if (r < 8) {
        int i = it * 16 + (hi ? 8 : 0) + j;
        // sigmas 6.4, 25.6, 51.2 -> 1/(2*s^2)
        const float inv2s2[3] = { 1.0f / 81.92f, 1.0f / 1310.72f, 1.0f / 5242.88f };
        float* op = out + ((size_t)(b * HH + i) * WW + w) * 6 + mask * 3;
        float q8[3];
#pragma unroll
        for (int s = 0; s < 3; ++s) {
            float v = __expf(-d2 * inv2s2[s]);                  // v_exp_f32
            q8[s] = fminf(fmaxf(rintf(v * 255.0f), 0.0f), 255.0f);
        }
        op[0] = q8[0]; op[1] = q8[1]; op[2] = q8[2];            // merged store
    }
}

// ---------------------------------------------------------------------------
extern "C" void kernel_launch(void* const* d_in, const int* in_sizes, int n_in,
                              void* d_out, int out_size, void* d_ws, size_t ws_size,
                              hipStream_t stream) {
    const int* tri = (const int*)d_in[0];   // trimap int32 [4,320,320,1]
    float* out = (float*)d_out;             // [4,320,320,6] f32
    float* gT  = (float*)d_ws;              // 2*4*320*320 f32 = 3.28 MB scratch

    edt_rows<<<dim3(2560 / 32), dim3(32), 0, stream>>>(tri, gT);
    edt_cols<<<dim3((2 * NB * IT * WW) / 8), dim3(256), 0, stream>>>(gT, out);
}